// SkipLastGNN_15272903704729
// MI455X (gfx1250) — compile-verified
//
#include <hip/hip_runtime.h>
#include <hip/hip_bf16.h>
#include <math.h>

typedef float v2f __attribute__((ext_vector_type(2)));
typedef float v8f __attribute__((ext_vector_type(8)));

// ---------------- utility kernels ----------------

__global__ void zero4_kernel(float4* __restrict__ p, long long n4) {
    long long i = (long long)blockIdx.x * blockDim.x + threadIdx.x;
    if (i < n4) p[i] = make_float4(0.f, 0.f, 0.f, 0.f);
}

__global__ void zero_kernel(float* __restrict__ p, long long n) {
    long long i = (long long)blockIdx.x * blockDim.x + threadIdx.x;
    if (i < n) p[i] = 0.0f;
}

__global__ void degree_kernel(const int* __restrict__ dst, float* __restrict__ deg, int E) {
    int e = blockIdx.x * blockDim.x + threadIdx.x;
    if (e < E) atomicAdd(&deg[dst[e]], 1.0f);
}

__global__ void invdeg_kernel(const float* __restrict__ deg, float* __restrict__ invdeg, int N_) {
    int i = blockIdx.x * blockDim.x + threadIdx.x;
    if (i < N_) invdeg[i] = 1.0f / fmaxf(deg[i], 1.0f);
}

// gates[i*3+j] = sigmoid(skip[i*3+j]) -- all 9, computed once
__global__ void gates_kernel(const float* __restrict__ skip, float* __restrict__ gates) {
    int t = threadIdx.x;
    if (t < 9) gates[t] = 1.0f / (1.0f + __expf(-skip[t]));
}

// gather: agg[dst, j*64 + q*4 .. +3] += gate[j] * allemb[j][src][q*4 .. +3]
// grid: x covers E*16 float4-quads, y = chunk j (0..nchunk-1)
__global__ __launch_bounds__(256)
void gather_kernel(const int* __restrict__ src, const int* __restrict__ dst,
                   const float* __restrict__ allemb, const float* __restrict__ gates,
                   float* __restrict__ agg, int N_, int dim, long long nquads) {
    long long tid = (long long)blockIdx.x * blockDim.x + threadIdx.x;
    if (tid >= nquads) return;
    int e = (int)(tid >> 4);
    int q = (int)(tid & 15);
    int j = blockIdx.y;
    float g = gates[j];
    const float4 v = *(const float4*)(allemb + ((size_t)j * N_ + src[e]) * 64 + q * 4);
    float* ap = agg + (size_t)dst[e] * dim + j * 64 + q * 4;
    atomicAdd(ap + 0, v.x * g);
    atomicAdd(ap + 1, v.y * g);
    atomicAdd(ap + 2, v.z * g);
    atomicAdd(ap + 3, v.w * g);
}

// ---------------- WMMA GEMM kernels (f32, 16x16x4) ----------------
// C[M,64] = act(A[M,K] @ B[K,64] + bias). One wave per 16x16 tile, 4 waves/block.
__global__ __launch_bounds__(128)
void gemm_bias_act_kernel(const float* __restrict__ A, const float* __restrict__ B,
                          const float* __restrict__ bias, float* __restrict__ C,
                          int K, int do_relu) {
    int wave = threadIdx.x >> 5;
    int lane = threadIdx.x & 31;
    int row0 = blockIdx.x * 16;
    int col0 = wave * 16;
    int m  = lane & 15;        // row within A-tile / col within B-tile
    int kh = lane >> 4;        // high half of K quad
    v8f c = {};
    for (int k = 0; k < K; k += 4) {
        int ka = k + 2 * kh;
        v2f a, b;
        const float* ap = A + (size_t)(row0 + m) * K + ka;
        a.x = ap[0];
        a.y = ap[1];
        b.x = B[(size_t)ka * 64 + col0 + m];
        b.y = B[(size_t)(ka + 1) * 64 + col0 + m];
        c = __builtin_amdgcn_wmma_f32_16x16x4_f32(false, a, false, b, (short)0, c, false, false);
    }
    float bb = bias[col0 + m];
    #pragma unroll
    for (int v = 0; v < 8; ++v) {
        float val = c[v] + bb;
        if (do_relu) val = fmaxf(val, 0.0f);
        C[(size_t)(row0 + v + 8 * kh) * 64 + col0 + m] = val;
    }
}

// SAGE layer: h = relu( (agg * inv_deg) @ wl + gated_emb @ wr + bl )
// gated_emb(m,k) = allemb[k/64][m][k%64] * gates[k/64]
__global__ __launch_bounds__(128)
void sage_gemm_kernel(const float* __restrict__ agg, const float* __restrict__ allemb,
                      const float* __restrict__ invdeg,
                      const float* __restrict__ wl, const float* __restrict__ bl,
                      const float* __restrict__ wr, const float* __restrict__ gatesrow,
                      float* __restrict__ out, int N_, int dim) {
    int wave = threadIdx.x >> 5;
    int lane = threadIdx.x & 31;
    int row0 = blockIdx.x * 16;
    int col0 = wave * 16;
    int m  = lane & 15;
    int kh = lane >> 4;
    float idg = invdeg[row0 + m];
    float gates[3];
    int nchunk = dim >> 6;
    for (int j = 0; j < nchunk; ++j) gates[j] = gatesrow[j];
    v8f c = {};
    // (agg * inv_deg) @ wl
    for (int k = 0; k < dim; k += 4) {
        int ka = k + 2 * kh;
        v2f a, b;
        const float* ap = agg + (size_t)(row0 + m) * dim + ka;
        a.x = ap[0] * idg;
        a.y = ap[1] * idg;
        b.x = wl[(size_t)ka * 64 + col0 + m];
        b.y = wl[(size_t)(ka + 1) * 64 + col0 + m];
        c = __builtin_amdgcn_wmma_f32_16x16x4_f32(false, a, false, b, (short)0, c, false, false);
    }
    // gated concat embeddings @ wr
    for (int k = 0; k < dim; k += 4) {
        int ka = k + 2 * kh;
        int j = ka >> 6, off = ka & 63;
        float g = gates[j];
        const float* ap = allemb + ((size_t)j * N_ + row0 + m) * 64 + off;
        v2f a, b;
        a.x = ap[0] * g;
        a.y = ap[1] * g;
        b.x = wr[(size_t)ka * 64 + col0 + m];
        b.y = wr[(size_t)(ka + 1) * 64 + col0 + m];
        c = __builtin_amdgcn_wmma_f32_16x16x4_f32(false, a, false, b, (short)0, c, false, false);
    }
    float bb = bl[col0 + m];
    #pragma unroll
    for (int v = 0; v < 8; ++v) {
        float val = fmaxf(c[v] + bb, 0.0f);
        out[(size_t)(row0 + v + 8 * kh) * 64 + col0 + m] = val;
    }
}

// ---------------- reduction + post MLP ----------------

// s[c] = sum over nodes of allemb[c/64][m][c%64]; one block per column (256 blocks)
__global__ __launch_bounds__(256)
void colsum_kernel(const float* __restrict__ allemb, float* __restrict__ s, int N_) {
    int c = blockIdx.x;
    int j = c >> 6, f = c & 63;
    const float* base = allemb + (size_t)j * N_ * 64 + f;
    float acc = 0.0f;
    for (int m2 = threadIdx.x; m2 < N_; m2 += blockDim.x) acc += base[(size_t)m2 * 64];
    __shared__ float red[256];
    red[threadIdx.x] = acc;
    __syncthreads();
    for (int sft = 128; sft > 0; sft >>= 1) {
        if (threadIdx.x < sft) red[threadIdx.x] += red[threadIdx.x + sft];
        __syncthreads();
    }
    if (threadIdx.x == 0) s[c] = red[0];
}

__global__ __launch_bounds__(256)
void post_kernel(const float* __restrict__ s,
                 const float* __restrict__ w1, const float* __restrict__ b1,
                 const float* __restrict__ w2, const float* __restrict__ b2,
                 const float* __restrict__ w3, const float* __restrict__ b3,
                 const float* __restrict__ w4, const float* __restrict__ b4,
                 float* __restrict__ out) {
    __shared__ float sv[256], h1[64], h2[64], h3[256];
    int t = threadIdx.x;
    sv[t] = s[t];
    __syncthreads();
    if (t < 64) {
        float acc = b1[t];
        for (int k = 0; k < 256; ++k) acc += sv[k] * w1[k * 64 + t];
        h1[t] = acc > 0.0f ? acc : 0.1f * acc;   // leaky_relu slope 0.1
    }
    __syncthreads();
    if (t < 64) {
        float acc = b2[t];
        for (int k = 0; k < 64; ++k) acc += h1[k] * w2[k * 64 + t];
        h2[t] = fmaxf(acc, 0.0f);
    }
    __syncthreads();
    {
        float acc = b3[t];
        for (int k = 0; k < 64; ++k) acc += h2[k] * w3[k * 256 + t];
        h3[t] = fmaxf(acc, 0.0f);
    }
    __syncthreads();
    if (t < 64) {
        float acc = b4[t];
        for (int k = 0; k < 256; ++k) acc += h3[k] * w4[k * 64 + t];
        out[t] = acc;
    }
}

// ---------------- launch ----------------

extern "C" void kernel_launch(void* const* d_in, const int* in_sizes, int n_in,
                              void* d_out, int out_size, void* d_ws, size_t ws_size,
                              hipStream_t stream) {
    const float* NF   = (const float*)d_in[0];
    const int*   eidx = (const int*)d_in[1];
    const float* pre_w1 = (const float*)d_in[2];
    const float* pre_b1 = (const float*)d_in[3];
    const float* pre_w2 = (const float*)d_in[4];
    const float* pre_b2 = (const float*)d_in[5];
    const float* skip   = (const float*)d_in[6];
    const float* wl[3] = {(const float*)d_in[7],  (const float*)d_in[10], (const float*)d_in[13]};
    const float* bl[3] = {(const float*)d_in[8],  (const float*)d_in[11], (const float*)d_in[14]};
    const float* wr[3] = {(const float*)d_in[9],  (const float*)d_in[12], (const float*)d_in[15]};
    const float* post_w1 = (const float*)d_in[16];
    const float* post_b1 = (const float*)d_in[17];
    const float* post_w2 = (const float*)d_in[18];
    const float* post_b2 = (const float*)d_in[19];
    const float* post_w3 = (const float*)d_in[20];
    const float* post_b3 = (const float*)d_in[21];
    const float* post_w4 = (const float*)d_in[22];
    const float* post_b4 = (const float*)d_in[23];

    const int N_ = in_sizes[0] / 128;   // 50000
    const int E_ = in_sizes[1] / 2;     // 800000
    const int* src = eidx;
    const int* dst = eidx + E_;

    float* ws = (float*)d_ws;
    float* allemb = ws;                               // [4][N][64]
    float* agg    = allemb + (size_t)4 * N_ * 64;     // [N][192] (also pre-MLP temp [N][64])
    float* deg    = agg + (size_t)N_ * 192;           // [N]
    float* invdeg = deg + N_;                         // [N]
    float* svec   = invdeg + N_;                      // [256]
    float* gates  = svec + 256;                       // [9]

    // degree + inverse degree + gates
    zero_kernel<<<(N_ + 255) / 256, 256, 0, stream>>>(deg, N_);
    degree_kernel<<<(E_ + 255) / 256, 256, 0, stream>>>(dst, deg, E_);
    invdeg_kernel<<<(N_ + 255) / 256, 256, 0, stream>>>(deg, invdeg, N_);
    gates_kernel<<<1, 16, 0, stream>>>(skip, gates);

    // pre-MLP: t1 = relu(NF @ pre_w1 + b1); x = t1 @ pre_w2 + b2 -> allemb chunk 0
    gemm_bias_act_kernel<<<N_ / 16, 128, 0, stream>>>(NF, pre_w1, pre_b1, agg, 128, 1);
    gemm_bias_act_kernel<<<N_ / 16, 128, 0, stream>>>(agg, pre_w2, pre_b2, allemb, 64, 0);

    // SAGE layers
    for (int i = 0; i < 3; ++i) {
        int dim = 64 * (i + 1);
        int nchunk = i + 1;
        long long aggn4 = (long long)N_ * dim / 4;
        zero4_kernel<<<(int)((aggn4 + 255) / 256), 256, 0, stream>>>((float4*)agg, aggn4);
        long long nquads = (long long)E_ * 16;  // per chunk: E edges * 16 float4s
        dim3 ggrid((unsigned)((nquads + 255) / 256), (unsigned)nchunk);
        gather_kernel<<<ggrid, 256, 0, stream>>>(src, dst, allemb, gates + i * 3,
                                                 agg, N_, dim, nquads);
        sage_gemm_kernel<<<N_ / 16, 128, 0, stream>>>(
            agg, allemb, invdeg, wl[i], bl[i], wr[i], gates + i * 3,
            allemb + (size_t)(i + 1) * N_ * 64, N_, dim);
    }

    // node sum + post MLP
    colsum_kernel<<<256, 256, 0, stream>>>(allemb, svec, N_);
    post_kernel<<<1, 256, 0, stream>>>(svec, post_w1, post_b1, post_w2, post_b2,
                                       post_w3, post_b3, post_w4, post_b4, (float*)d_out);
}